// ConvolutionBase_in_out_13073880449167
// MI455X (gfx1250) — compile-verified
//
#include <hip/hip_runtime.h>

#define N_NODES   100000
#define N_EDGES   1600000
#define D_FEAT    128
#define IN_CH     512
#define OUT_CH    256

typedef __attribute__((ext_vector_type(2))) float v2f;
typedef __attribute__((ext_vector_type(8))) float v8f;

// ---------------------------------------------------------------------------
// Kernel 1: zero the workspace (h accumulator + label sums + degree counters)
// ---------------------------------------------------------------------------
__global__ void zero_ws_kernel(float4* __restrict__ ws, size_t n4) {
    size_t i = (size_t)blockIdx.x * blockDim.x + threadIdx.x;
    size_t stride = (size_t)gridDim.x * blockDim.x;
    float4 z = make_float4(0.f, 0.f, 0.f, 0.f);
    for (; i < n4; i += stride) ws[i] = z;
}

// ---------------------------------------------------------------------------
// Kernel 2: one wave32 per edge. Lanes own 4 of the 128 features for the x
// gathers; the 4-float edge label is scattered RAW (opinion expansion is
// deferred: scatter_mean(label @ TW) == scatter_mean(label) @ TW).
//   h[row][  0..127] += x[col]          (out,   256 atomics/edge)
//   h[col][256..383] += x[row]          (inn)
//   lab_row[row][0..3] += label         (4 atomics/edge)
//   lab_col[col][0..3] += label         (4 atomics/edge)
// Native global_atomic_add_f32; accumulators are mostly L2-resident (192 MB).
// ---------------------------------------------------------------------------
__global__ void __launch_bounds__(256)
edge_scatter_kernel(const float* __restrict__ x,
                    const long long* __restrict__ ei,   // [2, E] int64
                    const float* __restrict__ elabel,   // [E, 4]
                    float* __restrict__ hacc,           // [N, 512]
                    float* __restrict__ lab_row,        // [N, 4]
                    float* __restrict__ lab_col,        // [N, 4]
                    float* __restrict__ cnt_row,
                    float* __restrict__ cnt_col) {
    unsigned wid  = (blockIdx.x * blockDim.x + threadIdx.x) >> 5;
    unsigned lane = threadIdx.x & 31;
    if (wid >= N_EDGES) return;

    long long row = ei[wid];
    long long col = ei[(size_t)N_EDGES + wid];
    int j0 = (int)lane * 4;

    float4 vc = *(const float4*)(x + (size_t)col * D_FEAT + j0); // x[col]
    float4 vr = *(const float4*)(x + (size_t)row * D_FEAT + j0); // x[row]

    float* hr = hacc + (size_t)row * IN_CH;
    float* hc = hacc + (size_t)col * IN_CH;

    unsafeAtomicAdd(hr + j0 + 0, vc.x);
    unsafeAtomicAdd(hr + j0 + 1, vc.y);
    unsafeAtomicAdd(hr + j0 + 2, vc.z);
    unsafeAtomicAdd(hr + j0 + 3, vc.w);

    unsafeAtomicAdd(hc + 256 + j0 + 0, vr.x);
    unsafeAtomicAdd(hc + 256 + j0 + 1, vr.y);
    unsafeAtomicAdd(hc + 256 + j0 + 2, vr.z);
    unsafeAtomicAdd(hc + 256 + j0 + 3, vr.w);

    if (lane < 4) {
        float lv = elabel[(size_t)wid * 4 + lane];
        unsafeAtomicAdd(lab_row + (size_t)row * 4 + lane, lv);
    } else if (lane < 8) {
        unsigned c = lane - 4;
        float lv = elabel[(size_t)wid * 4 + c];
        unsafeAtomicAdd(lab_col + (size_t)col * 4 + c, lv);
    } else if (lane == 8) {
        unsafeAtomicAdd(cnt_row + row, 1.0f);
    } else if (lane == 9) {
        unsafeAtomicAdd(cnt_col + col, 1.0f);
    }
}

// ---------------------------------------------------------------------------
// Kernel 3: expand label sums into the opinion segments of h (raw sums; the
// per-row mean scaling is folded into the GEMM A-fragment load):
//   h[n][128+j] = lab_row[n] . TW[:,j]     h[n][384+j] = lab_col[n] . TW[:,j]
// One wave per node, lane owns 4 of 128 features.
// ---------------------------------------------------------------------------
__global__ void __launch_bounds__(256)
fill_opinion_kernel(const float* __restrict__ lab_row,  // [N, 4]
                    const float* __restrict__ lab_col,  // [N, 4]
                    const float* __restrict__ tw,       // [4, 128]
                    float* __restrict__ hacc) {         // [N, 512]
    unsigned wid  = (blockIdx.x * blockDim.x + threadIdx.x) >> 5;
    unsigned lane = threadIdx.x & 31;
    if (wid >= N_NODES) return;
    int j0 = (int)lane * 4;

    float4 lr = *(const float4*)(lab_row + (size_t)wid * 4);
    float4 lc = *(const float4*)(lab_col + (size_t)wid * 4);
    float4 t0 = *(const float4*)(tw + 0 * D_FEAT + j0);
    float4 t1 = *(const float4*)(tw + 1 * D_FEAT + j0);
    float4 t2 = *(const float4*)(tw + 2 * D_FEAT + j0);
    float4 t3 = *(const float4*)(tw + 3 * D_FEAT + j0);

    float4 er, ec;
    er.x = lr.x * t0.x + lr.y * t1.x + lr.z * t2.x + lr.w * t3.x;
    er.y = lr.x * t0.y + lr.y * t1.y + lr.z * t2.y + lr.w * t3.y;
    er.z = lr.x * t0.z + lr.y * t1.z + lr.z * t2.z + lr.w * t3.z;
    er.w = lr.x * t0.w + lr.y * t1.w + lr.z * t2.w + lr.w * t3.w;
    ec.x = lc.x * t0.x + lc.y * t1.x + lc.z * t2.x + lc.w * t3.x;
    ec.y = lc.x * t0.y + lc.y * t1.y + lc.z * t2.y + lc.w * t3.y;
    ec.z = lc.x * t0.z + lc.y * t1.z + lc.z * t2.z + lc.w * t3.z;
    ec.w = lc.x * t0.w + lc.y * t1.w + lc.z * t2.w + lc.w * t3.w;

    float* hrow = hacc + (size_t)wid * IN_CH;
    *(float4*)(hrow + 128 + j0) = er;
    *(float4*)(hrow + 384 + j0) = ec;
}

// ---------------------------------------------------------------------------
// Kernel 4: counts -> reciprocals (1 / max(cnt, 1))
// ---------------------------------------------------------------------------
__global__ void invert_counts_kernel(float* __restrict__ cnt, int n) {
    int i = blockIdx.x * blockDim.x + threadIdx.x;
    if (i < n) {
        float c = cnt[i];
        cnt[i] = 1.0f / fmaxf(c, 1.0f);
    }
}

// ---------------------------------------------------------------------------
// Kernel 5: out[N,256] = diag-normalize(h)[N,512] @ W[512,256] + bias
// fp32 WMMA: V_WMMA_F32_16X16X4_F32. Wave register-blocks 2 M-tiles x
// 2 N-tiles (4 accumulators, 32 VGPRs): 6 vmem loads per 4 WMMAs with 2x
// A- and B-fragment reuse. Block = 256 threads (8 waves) = 32 rows x 256
// cols; N_NODES/32 = 3125 exact. Mean normalization folded into A load
// (inv_row for k<256, inv_col for k>=256).
// ---------------------------------------------------------------------------
__global__ void __launch_bounds__(256)
gemm_wmma_f32_kernel(const float* __restrict__ h,       // [N, 512]
                     const float* __restrict__ w,       // [512, 256]
                     const float* __restrict__ bias,    // [256]
                     const float* __restrict__ inv_row, // [N]
                     const float* __restrict__ inv_col, // [N]
                     float* __restrict__ out) {         // [N, 256]
    const int lane  = threadIdx.x & 31;
    const int wave  = threadIdx.x >> 5;
    const int row0  = blockIdx.x * 32;
    const int m     = lane & 15;
    const int kgrp  = (lane >> 4) * 2;          // 0 for lanes 0-15, 2 for 16-31
    const int arow0 = row0 + m;                 // M-tile 0 row for this lane
    const int arow1 = row0 + 16 + m;            // M-tile 1 row
    const int nb    = wave * 32 + m;            // N-tile 0 column for this lane

    const float ir0 = inv_row[arow0], ic0 = inv_col[arow0];
    const float ir1 = inv_row[arow1], ic1 = inv_col[arow1];

    v8f c00 = (v8f){}, c01 = (v8f){}, c10 = (v8f){}, c11 = (v8f){};

    const float* h0 = h + (size_t)arow0 * IN_CH;
    const float* h1 = h + (size_t)arow1 * IN_CH;

    for (int k = 0; k < IN_CH; k += 4) {
        const int kk = k + kgrp;                // stays within same 256-half
        float2 a0v = *(const float2*)(h0 + kk);
        float2 a1v = *(const float2*)(h1 + kk);
        const float i0 = (k < 256) ? ir0 : ic0;
        const float i1 = (k < 256) ? ir1 : ic1;
        v2f a0, a1;
        a0[0] = a0v.x * i0; a0[1] = a0v.y * i0;
        a1[0] = a1v.x * i1; a1[1] = a1v.y * i1;

        const float* w0 = w + (size_t)kk * OUT_CH;        // row kk
        const float* w1 = w0 + OUT_CH;                    // row kk+1
        v2f b0, b1;
        b0[0] = w0[nb];      b0[1] = w1[nb];
        b1[0] = w0[nb + 16]; b1[1] = w1[nb + 16];

        c00 = __builtin_amdgcn_wmma_f32_16x16x4_f32(false, a0, false, b0, (short)0, c00, false, false);
        c01 = __builtin_amdgcn_wmma_f32_16x16x4_f32(false, a0, false, b1, (short)0, c01, false, false);
        c10 = __builtin_amdgcn_wmma_f32_16x16x4_f32(false, a1, false, b0, (short)0, c10, false, false);
        c11 = __builtin_amdgcn_wmma_f32_16x16x4_f32(false, a1, false, b1, (short)0, c11, false, false);
    }

    // D layout: lanes 0-15 -> rows base+v, lanes 16-31 -> rows base+8+v
    const int og   = (lane >> 4) * 8;
    const float b0v = bias[nb];
    const float b1v = bias[nb + 16];
#pragma unroll
    for (int v = 0; v < 8; ++v) {
        const size_t r0 = (size_t)(row0 + og + v) * OUT_CH;
        const size_t r1 = (size_t)(row0 + 16 + og + v) * OUT_CH;
        out[r0 + nb]      = c00[v] + b0v;
        out[r0 + nb + 16] = c01[v] + b1v;
        out[r1 + nb]      = c10[v] + b0v;
        out[r1 + nb + 16] = c11[v] + b1v;
    }
}

// ---------------------------------------------------------------------------
extern "C" void kernel_launch(void* const* d_in, const int* in_sizes, int n_in,
                              void* d_out, int out_size, void* d_ws, size_t ws_size,
                              hipStream_t stream) {
    const float*     x      = (const float*)d_in[0];        // [N, 128]
    const long long* ei     = (const long long*)d_in[1];    // [2, E] int64
    const float*     elabel = (const float*)d_in[2];        // [E, 4]
    const float*     weight = (const float*)d_in[3];        // [512, 256]
    const float*     tw     = (const float*)d_in[4];        // [4, 128]
    const float*     bias   = (const float*)d_in[5];        // [256]
    float*           outp   = (float*)d_out;                // [N, 256]

    // Workspace: h[N*512] | lab_row[N*4] | lab_col[N*4] | cnt_row[N] | cnt_col[N]
    const size_t n_h    = (size_t)N_NODES * IN_CH;
    const size_t n_tot  = n_h + 8 * (size_t)N_NODES + 2 * (size_t)N_NODES;
    const size_t need   = n_tot * sizeof(float);
    if (ws_size < need) return;  // insufficient scratch; deterministic no-op

    float* hacc    = (float*)d_ws;
    float* lab_row = hacc + n_h;
    float* lab_col = lab_row + (size_t)N_NODES * 4;
    float* cnt_row = lab_col + (size_t)N_NODES * 4;
    float* cnt_col = cnt_row + N_NODES;

    // 1. zero accumulators (total float count divisible by 4)
    zero_ws_kernel<<<8192, 256, 0, stream>>>((float4*)d_ws, n_tot / 4);

    // 2. edge scatter: one wave per edge -> E/8 blocks of 256 threads
    const int edge_blocks = (N_EDGES * 32 + 255) / 256;   // 200000
    edge_scatter_kernel<<<edge_blocks, 256, 0, stream>>>(
        x, ei, elabel, hacc, lab_row, lab_col, cnt_row, cnt_col);

    // 3. expand label sums -> opinion segments of h (one wave per node)
    const int fill_blocks = (N_NODES * 32 + 255) / 256;   // 12500
    fill_opinion_kernel<<<fill_blocks, 256, 0, stream>>>(
        lab_row, lab_col, tw, hacc);

    // 4. counts -> reciprocals (both arrays contiguous: 2N elements)
    const int nc = 2 * N_NODES;
    invert_counts_kernel<<<(nc + 255) / 256, 256, 0, stream>>>(cnt_row, nc);

    // 5. fp32 WMMA GEMM with fused mean-normalization + bias
    gemm_wmma_f32_kernel<<<N_NODES / 32, 256, 0, stream>>>(
        hacc, weight, bias, cnt_row, cnt_col, outp);

    (void)in_sizes; (void)n_in; (void)out_size;
}